// Pretrain_51453708206795
// MI455X (gfx1250) — compile-verified
//
#include <hip/hip_runtime.h>
#include <math.h>

#define NPIX   (4*112*112)   // 50176
#define DDIM   512
#define KCLS   4
#define MPROTO 10
#define NCOL   (KCLS*MPROTO) // 40
#define NCOLP  48            // padded to 3 WMMA column tiles
#define HWSZ   (112*112)
#define GAMMA  0.999f
#define LN_EPS 1e-5f
#define INV_SK_EPS 20.0f     // 1/0.05

typedef float v2f __attribute__((ext_vector_type(2)));
typedef float v8f __attribute__((ext_vector_type(8)));

__device__ inline float block_reduce_sum(float v, float* sm) {
  for (int o = 16; o > 0; o >>= 1) v += __shfl_down(v, o, 32);
  int lane = threadIdx.x & 31, wid = threadIdx.x >> 5;
  int nw = (blockDim.x + 31) >> 5;
  if (lane == 0) sm[wid] = v;
  __syncthreads();
  float s = (threadIdx.x < nw) ? sm[threadIdx.x] : 0.0f;
  if (wid == 0) {
    for (int o = 16; o > 0; o >>= 1) s += __shfl_down(s, o, 32);
    if (lane == 0) sm[0] = s;
  }
  __syncthreads();
  float r = sm[0];
  __syncthreads();
  return r;
}

// _c = l2_normalize(layer_norm(x)) ; one block (256 thr) per row of 512
__global__ void k_cnorm(const float* __restrict__ x, const float* __restrict__ g,
                        const float* __restrict__ b, float* __restrict__ c) {
  __shared__ float sm[16];
  int n = blockIdx.x, t = threadIdx.x;
  float x0 = x[(size_t)n*DDIM + t], x1 = x[(size_t)n*DDIM + t + 256];
  float mu = block_reduce_sum(x0 + x1, sm) * (1.0f/DDIM);
  float d0 = x0 - mu, d1 = x1 - mu;
  float var = block_reduce_sum(d0*d0 + d1*d1, sm) * (1.0f/DDIM);
  float inv = rsqrtf(var + LN_EPS);
  float y0 = d0*inv*g[t] + b[t];
  float y1 = d1*inv*g[t+256] + b[t+256];
  float nn = block_reduce_sum(y0*y0 + y1*y1, sm);
  float r = 1.0f / fmaxf(sqrtf(nn), 1e-12f);
  c[(size_t)n*DDIM + t]       = y0*r;
  c[(size_t)n*DDIM + t + 256] = y1*r;
}

// normalized prototypes, zero-padded rows [40,48)
__global__ void k_pnorm(const float* __restrict__ p, float* __restrict__ pn) {
  __shared__ float sm[16];
  int row = blockIdx.x, t = threadIdx.x;
  float x0 = 0.f, x1 = 0.f;
  if (row < NCOL) { x0 = p[(size_t)row*DDIM + t]; x1 = p[(size_t)row*DDIM + t + 256]; }
  float nn = block_reduce_sum(x0*x0 + x1*x1, sm);
  float r = 1.0f / fmaxf(sqrtf(nn), 1e-12f);
  pn[(size_t)row*DDIM + t]       = x0*r;
  pn[(size_t)row*DDIM + t + 256] = x1*r;
}

// masks[n][km] = _c[n,:] . pn[km,:]  via V_WMMA_F32_16X16X4_F32
// one wave per 16-pixel tile; A fragment shared across 3 column-tile accumulators
__global__ void k_gemm(const float* __restrict__ c, const float* __restrict__ pn,
                       float* __restrict__ masks) {
  int lane = threadIdx.x;
  int n0 = blockIdx.x * 16;
  int r    = lane & 15;
  int koff = (lane >> 4) * 2;           // A/B: VGPR0=K0/K1 (lanes 0-15), K2/K3 (lanes 16-31)
  const float* arow  = c  + (size_t)(n0 + r)*DDIM + koff;
  const float* brow0 = pn + (size_t)(r      )*DDIM + koff;   // cols  0..15
  const float* brow1 = pn + (size_t)(r + 16 )*DDIM + koff;   // cols 16..31
  const float* brow2 = pn + (size_t)(r + 32 )*DDIM + koff;   // cols 32..47 (40..47 zero)
  v8f acc0 = {}, acc1 = {}, acc2 = {};
  for (int d = 0; d < DDIM; d += 4) {
    v2f a, b0, b1, b2;
    a.x  = arow[d];  a.y  = arow[d + 1];
    b0.x = brow0[d]; b0.y = brow0[d + 1];
    b1.x = brow1[d]; b1.y = brow1[d + 1];
    b2.x = brow2[d]; b2.y = brow2[d + 1];
    acc0 = __builtin_amdgcn_wmma_f32_16x16x4_f32(false, a, false, b0, (short)0, acc0,
                                                 false, false);
    acc1 = __builtin_amdgcn_wmma_f32_16x16x4_f32(false, a, false, b1, (short)0, acc1,
                                                 false, false);
    acc2 = __builtin_amdgcn_wmma_f32_16x16x4_f32(false, a, false, b2, (short)0, acc2,
                                                 false, false);
  }
  int rowbase = (lane >> 4) * 8;        // C layout: VGPR i -> M=i (lo lanes), M=8+i (hi lanes)
  int coln = lane & 15;
  for (int i = 0; i < 8; i++) {
    size_t rb = (size_t)(n0 + rowbase + i)*NCOLP + coln;
    masks[rb     ] = acc0[i];
    masks[rb + 16] = acc1[i];
    masks[rb + 32] = acc2[i];
  }
}

// out_seg = layer_norm(max_m masks) ; pred/correct
__global__ void k_outseg(const float* __restrict__ masks, const float* __restrict__ mg,
                         const float* __restrict__ mb, const int* __restrict__ gt,
                         float* __restrict__ out, int* __restrict__ correct) {
  int n = blockIdx.x*blockDim.x + threadIdx.x;
  if (n >= NPIX) return;
  float o[KCLS];
  for (int k = 0; k < KCLS; k++) {
    float mx = -INFINITY;
    for (int m = 0; m < MPROTO; m++)
      mx = fmaxf(mx, masks[(size_t)n*NCOLP + k*MPROTO + m]);
    o[k] = mx;
  }
  float mu = 0.f;
  for (int k = 0; k < KCLS; k++) mu += o[k];
  mu *= 0.25f;
  float var = 0.f;
  for (int k = 0; k < KCLS; k++) { float d = o[k]-mu; var += d*d; }
  var *= 0.25f;
  float inv = rsqrtf(var + LN_EPS);
  int b = n / HWSZ, hw = n % HWSZ;
  float best = -INFINITY; int pred = 0;
  for (int k = 0; k < KCLS; k++) {
    float y = (o[k]-mu)*inv*mg[k] + mb[k];
    out[(size_t)b*KCLS*HWSZ + (size_t)k*HWSZ + hw] = y;
    if (y > best) { best = y; pred = k; }
  }
  correct[n] = (gt[n] == pred) ? 1 : 0;
}

// Sinkhorn row-sum pass: S[km] = sum_n E[n,m]*b_n ; pass0 also counts pixels per class
__global__ void k_spass(const float* __restrict__ masks, const int* __restrict__ gt,
                        const float* __restrict__ a, const float* __restrict__ safecnt,
                        float* __restrict__ S, float* __restrict__ cnt, int pass) {
  __shared__ float Sl[NCOL];
  __shared__ float Cl[KCLS];
  int t = threadIdx.x;
  if (t < NCOL) Sl[t] = 0.f;
  if (t < KCLS) Cl[t] = 0.f;
  __syncthreads();
  for (int n = blockIdx.x*blockDim.x + t; n < NPIX; n += gridDim.x*blockDim.x) {
    int k = gt[n];
    const float* mrow = masks + (size_t)n*NCOLP + k*MPROTO;
    float e[MPROTO];
    for (int m = 0; m < MPROTO; m++) e[m] = expf(mrow[m] * INV_SK_EPS);
    float bfac = 1.0f;
    if (pass != 0) {
      float T = 0.f;
      for (int m = 0; m < MPROTO; m++) T += e[m]*a[k*MPROTO+m];
      bfac = 1.0f / (T * safecnt[k]);   // col>0 always holds for sel pixels
    }
    for (int m = 0; m < MPROTO; m++) atomicAdd(&Sl[k*MPROTO+m], e[m]*bfac);
    if (pass == 0) atomicAdd(&Cl[k], 1.0f);
  }
  __syncthreads();
  if (t < NCOL) atomicAdd(&S[t], Sl[t]);
  if (pass == 0 && t < KCLS) atomicAdd(&cnt[t], Cl[t]);
}

// a_m <- a_m / (max(a_m*S_m,1e-12)*M) ; pass0 initializes a=1/max(total,1e-12), safecnt
__global__ void k_update(const float* __restrict__ S, float* __restrict__ a,
                         const float* __restrict__ cnt, float* __restrict__ safecnt,
                         int pass) {
  int t = threadIdx.x;
  if (pass == 0 && t < KCLS) safecnt[t] = fmaxf(cnt[t], 1.0f);
  if (t < NCOL) {
    int k = t / MPROTO;
    float av;
    if (pass == 0) {
      float total = 0.f;
      for (int m = 0; m < MPROTO; m++) total += S[k*MPROTO+m];
      av = 1.0f / fmaxf(total, 1e-12f);
    } else {
      av = a[t];
    }
    float rs = av * S[t];
    a[t] = av / (fmaxf(rs, 1e-12f) * (float)MPROTO);
  }
}

// hard assignment: argmax_m E[n,m]*a_m (per-pixel b_n > 0 cancels); gated by correct
__global__ void k_assign(const float* __restrict__ masks, const int* __restrict__ gt,
                         const float* __restrict__ a, const int* __restrict__ correct,
                         int* __restrict__ assign) {
  int n = blockIdx.x*blockDim.x + threadIdx.x;
  if (n >= NPIX) return;
  int k = gt[n];
  const float* mrow = masks + (size_t)n*NCOLP + k*MPROTO;
  float best = -INFINITY; int bm = 0;
  for (int m = 0; m < MPROTO; m++) {
    float v = expf(mrow[m] * INV_SK_EPS) * a[k*MPROTO+m];
    if (v > best) { best = v; bm = m; }   // strict > keeps first max (jnp.argmax)
  }
  assign[n] = correct[n] ? (k*MPROTO + bm) : -1;
}

// f[km][d] = sum over assigned pixels of _c[n][d] ; deterministic gather per (km)
__global__ void k_faccum(const float* __restrict__ c, const int* __restrict__ assign,
                         float* __restrict__ f, float* __restrict__ ncnt) {
  int km = blockIdx.x, d = threadIdx.x;  // 512 threads
  float acc = 0.f; int cm = 0;
  for (int n = 0; n < NPIX; n++) {
    if (assign[n] == km) { acc += c[(size_t)n*DDIM + d]; cm++; }
  }
  f[(size_t)km*DDIM + d] = acc;
  if (d == 0) ncnt[km] = (float)cm;
}

// normalize f, sims, momentum update, final renormalize ; one block per class
__global__ void k_final(const float* __restrict__ f, const float* __restrict__ ncnt,
                        const float* __restrict__ pn, float* __restrict__ outp,
                        float* __restrict__ sims) {
  __shared__ float sm[16];
  int k = blockIdx.x, d = threadIdx.x;   // 512 threads
  float totn = 0.f;
  for (int m = 0; m < MPROTO; m++) totn += ncnt[k*MPROTO+m];
  float anyh = (totn > 0.f) ? 1.f : 0.f;
  float fn[MPROTO];
  float sumself = 0.f, gsum = 0.f;
  for (int m = 0; m < MPROTO; m++) {
    float fv = f[(size_t)(k*MPROTO+m)*DDIM + d];
    float nn = block_reduce_sum(fv*fv, sm);
    float r = 1.0f / fmaxf(sqrtf(nn), 1e-12f);
    fn[m] = fv * r;
    sumself += fn[m]*fn[m];
    gsum += fn[m];
  }
  float g2 = block_reduce_sum(gsum*gsum, sm);
  float ss = block_reduce_sum(sumself, sm);
  if (d == 0) sims[k] = 0.5f*(g2 - ss)*anyh;   // sum of upper-triangle dot products
  for (int m = 0; m < MPROTO; m++) {
    float p = pn[(size_t)(k*MPROTO+m)*DDIM + d];
    float upd = p*GAMMA + fn[m]*(1.0f - GAMMA);
    float nv = (ncnt[k*MPROTO+m] != 0.f && anyh > 0.f) ? upd : p;
    float nn = block_reduce_sum(nv*nv, sm);
    float r = 1.0f / fmaxf(sqrtf(nn), 1e-12f);
    outp[(size_t)(k*MPROTO+m)*DDIM + d] = nv*r;
  }
}

extern "C" void kernel_launch(void* const* d_in, const int* in_sizes, int n_in,
                              void* d_out, int out_size, void* d_ws, size_t ws_size,
                              hipStream_t stream) {
  (void)in_sizes; (void)n_in; (void)out_size; (void)ws_size;
  const float* x     = (const float*)d_in[0];
  const float* ln_g  = (const float*)d_in[1];
  const float* ln_b  = (const float*)d_in[2];
  const float* mn_g  = (const float*)d_in[3];
  const float* mn_b  = (const float*)d_in[4];
  const float* proto = (const float*)d_in[5];
  const int*   gt    = (const int*)d_in[6];

  float* out   = (float*)d_out;                 // [B,K,H,W] = 200704 floats
  float* outp  = out + (size_t)NPIX*KCLS;       // new_protos [K,M,D] = 20480
  float* sims  = outp + (size_t)NCOL*DDIM;      // [K] = 4

  float* ws = (float*)d_ws;
  size_t off = 0;
  float* C   = ws + off; off += (size_t)NPIX*DDIM;   // _c
  float* MK  = ws + off; off += (size_t)NPIX*NCOLP;  // masks (padded)
  float* PN  = ws + off; off += (size_t)NCOLP*DDIM;  // normalized prototypes
  float* F   = ws + off; off += (size_t)NCOL*DDIM;   // f
  float* S   = ws + off; off += NCOL;                // sinkhorn row sums
  float* A   = ws + off; off += NCOL;                // sinkhorn row scaling a_m
  float* CNT = ws + off; off += KCLS;                // per-class pixel count
  float* SC  = ws + off; off += KCLS;                // safe_cnt
  float* NC  = ws + off; off += NCOL;                // per-prototype counts n
  int* CORR  = (int*)(ws + off); off += NPIX;        // correct flags
  int* ASG   = (int*)(ws + off); off += NPIX;        // per-pixel assignment

  k_cnorm<<<NPIX, 256, 0, stream>>>(x, ln_g, ln_b, C);
  k_pnorm<<<NCOLP, 256, 0, stream>>>(proto, PN);

  k_gemm<<<NPIX/16, 32, 0, stream>>>(C, PN, MK);     // 3136 waves, 3 col tiles each

  k_outseg<<<(NPIX+255)/256, 256, 0, stream>>>(MK, mn_g, mn_b, gt, out, CORR);

  for (int pass = 0; pass < 3; pass++) {
    hipMemsetAsync(S, 0, NCOL*sizeof(float), stream);
    if (pass == 0) hipMemsetAsync(CNT, 0, KCLS*sizeof(float), stream);
    k_spass<<<196, 256, 0, stream>>>(MK, gt, A, SC, S, CNT, pass);
    k_update<<<1, 64, 0, stream>>>(S, A, CNT, SC, pass);
  }

  k_assign<<<(NPIX+255)/256, 256, 0, stream>>>(MK, gt, A, CORR, ASG);
  k_faccum<<<NCOL, DDIM, 0, stream>>>(C, ASG, F, NC);
  k_final<<<KCLS, DDIM, 0, stream>>>(F, NC, PN, outp, sims);
}